// LGCN_Encoder_52493090292053
// MI455X (gfx1250) — compile-verified
//
#include <hip/hip_runtime.h>
#include <stdint.h>

#define USER_NUM 100000
#define ITEM_NUM 200000
#define N_NODES  (USER_NUM + ITEM_NUM)
#define N_EDGES  1250000
#define EMB      64
#define EPB      512          // edges staged per block (6 KB LDS)
#define NVEC     (N_NODES * (EMB / 4))   // 4,800,000 float4s per [N,64] buffer

// ---------------------------------------------------------------------------
// zero both ping-pong buffers (laid out contiguously in d_ws)
// ---------------------------------------------------------------------------
__global__ __launch_bounds__(256) void lgcn_zero2(float4* __restrict__ bufs) {
    const size_t i = (size_t)blockIdx.x * blockDim.x + threadIdx.x;
    if (i >= (size_t)2 * NVEC) return;
    bufs[i] = make_float4(0.f, 0.f, 0.f, 0.f);
}

// ---------------------------------------------------------------------------
// spmm: y[row,:] += val * x[col,:] over a 512-edge chunk per block.
//  - gather source is split: col < split -> srcA[col,:], else srcB[col-split,:]
//    (pass srcA=srcB=buf, split=0-equivalent by srcB = srcA + split rows)
//  - edge lists staged to LDS with global_load_async_to_lds_b64 (ASYNCcnt)
//  - each 16-lane half-wave handles one edge: B128 gather + 4 f32 atomics/lane
// ---------------------------------------------------------------------------
__global__ __launch_bounds__(256) void lgcn_spmm(const int*   __restrict__ rows,
                                                 const int*   __restrict__ cols,
                                                 const float* __restrict__ vals,
                                                 const float4* __restrict__ srcA,
                                                 const float4* __restrict__ srcB,
                                                 float*       __restrict__ y) {
    __shared__ __align__(16) int   sRows[EPB];
    __shared__ __align__(16) int   sCols[EPB];
    __shared__ __align__(16) float sVals[EPB];

    const int chunk = blockIdx.x * EPB;
    const int cnt   = min(EPB, N_EDGES - chunk);   // always even
    const int t     = threadIdx.x;

    // --- async stage: thread t copies bytes [8t, 8t+8) of each edge array ---
    const uint32_t byteOff = (uint32_t)t * 8u;
    if (byteOff < (uint32_t)cnt * 4u) {
        const uint32_t ldsR = (uint32_t)(uintptr_t)(&sRows[0]) + byteOff;
        const uint32_t ldsC = (uint32_t)(uintptr_t)(&sCols[0]) + byteOff;
        const uint32_t ldsV = (uint32_t)(uintptr_t)(&sVals[0]) + byteOff;
        const void* gR = (const void*)(rows + chunk);
        const void* gC = (const void*)(cols + chunk);
        const void* gV = (const void*)(vals + chunk);
        asm volatile(
            "global_load_async_to_lds_b64 %0, %3, %4\n\t"
            "global_load_async_to_lds_b64 %1, %3, %5\n\t"
            "global_load_async_to_lds_b64 %2, %3, %6"
            :
            : "v"(ldsR), "v"(ldsC), "v"(ldsV), "v"(byteOff),
              "s"(gR), "s"(gC), "s"(gV)
            : "memory");
    }
    asm volatile("s_wait_asynccnt 0" ::: "memory");
    __syncthreads();

    // --- process: half-wave (16 lanes) per edge, lane handles 4 floats ---
    const int lane = t & 15;
    for (int e = (t >> 4); e < cnt; e += 16) {
        const int   row = sRows[e];
        const int   col = sCols[e];
        const float w   = sVals[e];
        // uniform select within the half-wave (same col for all 16 lanes)
        const float4* src = (col < USER_NUM)
                          ? (srcA + (size_t)col * (EMB / 4))
                          : (srcB + (size_t)(col - USER_NUM) * (EMB / 4));
        float4 v = src[lane];
        float* dst = y + (size_t)row * EMB + lane * 4;
        unsafeAtomicAdd(dst + 0, v.x * w);
        unsafeAtomicAdd(dst + 1, v.y * w);
        unsafeAtomicAdd(dst + 2, v.z * w);
        unsafeAtomicAdd(dst + 3, v.w * w);
    }
}

// ---------------------------------------------------------------------------
// k1: out = ego(user,item) + layer1; layer1buf = 0 (re-used as layer3 target)
// ---------------------------------------------------------------------------
__global__ __launch_bounds__(256) void lgcn_ego_add_clear(const float4* __restrict__ user,
                                                          const float4* __restrict__ item,
                                                          const float4* __restrict__ e1,
                                                          float4* __restrict__ out,
                                                          float4* __restrict__ clearbuf) {
    const size_t i = (size_t)blockIdx.x * blockDim.x + threadIdx.x;
    const size_t userEnd = (size_t)USER_NUM * (EMB / 4);
    if (i >= (size_t)NVEC) return;
    float4 g = (i < userEnd) ? user[i] : item[i - userEnd];
    float4 l = e1[i];
    g.x += l.x; g.y += l.y; g.z += l.z; g.w += l.w;
    out[i] = g;
    clearbuf[i] = make_float4(0.f, 0.f, 0.f, 0.f);
}

// ---------------------------------------------------------------------------
// final: out = (out + e2 + e3) * 0.25
// ---------------------------------------------------------------------------
__global__ __launch_bounds__(256) void lgcn_final(float4* __restrict__ out,
                                                  const float4* __restrict__ e2,
                                                  const float4* __restrict__ e3) {
    const size_t i = (size_t)blockIdx.x * blockDim.x + threadIdx.x;
    if (i >= (size_t)NVEC) return;
    float4 a = out[i];
    float4 b = e2[i];
    float4 c = e3[i];
    a.x = (a.x + b.x + c.x) * 0.25f;
    a.y = (a.y + b.y + c.y) * 0.25f;
    a.z = (a.z + b.z + c.z) * 0.25f;
    a.w = (a.w + b.w + c.w) * 0.25f;
    out[i] = a;
}

// ---------------------------------------------------------------------------
extern "C" void kernel_launch(void* const* d_in, const int* in_sizes, int n_in,
                              void* d_out, int out_size, void* d_ws, size_t ws_size,
                              hipStream_t stream) {
    const float4* user = (const float4*)d_in[0];   // [USER_NUM, 64]
    const float4* item = (const float4*)d_in[1];   // [ITEM_NUM, 64]
    const float*  vals = (const float*)d_in[2];    // [N_EDGES]
    const int*    rows = (const int*)  d_in[3];    // [N_EDGES]
    const int*    cols = (const int*)  d_in[4];    // [N_EDGES]

    float4* out  = (float4*)d_out;                         // [N_NODES, 64]
    float4* bufA = (float4*)d_ws;                          // ping (e1, then e3)
    float4* bufB = bufA + (size_t)NVEC;                    // pong (e2)

    const int vecBlocks  = (NVEC + 255) / 256;             // 18750
    const int zeroBlocks = (2 * NVEC + 255) / 256;
    const int spmmBlocks = (N_EDGES + EPB - 1) / EPB;      // 2442

    // zero both scatter targets
    lgcn_zero2<<<zeroBlocks, 256, 0, stream>>>(bufA);

    // layer 1: gather directly from (user|item) -> bufA
    lgcn_spmm<<<spmmBlocks, 256, 0, stream>>>(rows, cols, vals,
                                              user, item, (float*)bufA);
    // layer 2: gather bufA (unified [N,64]) -> bufB   (runs before bufA is cleared)
    lgcn_spmm<<<spmmBlocks, 256, 0, stream>>>(rows, cols, vals,
                                              bufA, bufA + (size_t)USER_NUM * (EMB / 4),
                                              (float*)bufB);
    // out = ego + e1 ; bufA = 0 (becomes layer-3 scatter target)
    lgcn_ego_add_clear<<<vecBlocks, 256, 0, stream>>>(user, item, bufA, out, bufA);

    // layer 3: gather bufB -> bufA
    lgcn_spmm<<<spmmBlocks, 256, 0, stream>>>(rows, cols, vals,
                                              bufB, bufB + (size_t)USER_NUM * (EMB / 4),
                                              (float*)bufA);
    // out = (out + e2 + e3) / 4
    lgcn_final<<<vecBlocks, 256, 0, stream>>>(out, bufB, bufA);
}